// VanillaRNN_16527034155502
// MI455X (gfx1250) — compile-verified
//
#include <hip/hip_runtime.h>
#include <math.h>

typedef __attribute__((ext_vector_type(2))) float v2f;
typedef __attribute__((ext_vector_type(8))) float v8f;

#define B_ 128
#define T_ 512
#define I_ 128
#define H_ 2048
#define O_ 10

#define KCHUNK 64
#define LDS_PITCH 68   // pitch%64==4 -> A-frag reads hit all 64 banks conflict-free

// ---------------------------------------------------------------------------
// Weight repack into WMMA-native B layout:
//   Wt[k*N + 4*n + j] = W[(k&~3) + j][n],  j = 0..3  (K-groups of 4 per column)
// -> a lane's B fragment {W[k+2lh][n], W[k+2lh+1][n]} is one aligned b64,
//    and a wave's 32 fragments form one contiguous 256B block.
// ---------------------------------------------------------------------------
__global__ void repack_kernel(const float* __restrict__ W,  // [K,N] row-major
                              float* __restrict__ Wt,       // [K/4, N, 4]
                              int K, int N)
{
    const int g = blockIdx.x * blockDim.x + threadIdx.x;   // group = (k0, n)
    const int total = (K >> 2) * N;
    if (g >= total) return;
    const int k0 = g / N;
    const int n  = g - k0 * N;
    float4 v;
    v.x = W[(size_t)(4 * k0 + 0) * N + n];
    v.y = W[(size_t)(4 * k0 + 1) * N + n];
    v.z = W[(size_t)(4 * k0 + 2) * N + n];
    v.w = W[(size_t)(4 * k0 + 3) * N + n];
    ((float4*)Wt)[g] = v;                                   // fully coalesced
}

// ---------------------------------------------------------------------------
// One RNN timestep: h_out = tanh(x[:,t,:] @ W_hx + h_in @ W_hh + b_h)
// Block = 128 threads (4 waves). Block tile: 32(M) x 64(N).
// Wave tile: 32x16 = two 16x16 accumulators sharing each B fragment.
// grid = (H_/64, B_/32) = (32, 4) = 128 blocks, 512 waves.
// PACKED: B fragment = one global_load_b64 from repacked weights.
// ---------------------------------------------------------------------------
template <bool PACKED>
__global__ __launch_bounds__(128)
void rnn_step_kernel(const float* __restrict__ x,      // [B,T,I]
                     const float* __restrict__ Wx,     // W_hx (raw or packed)
                     const float* __restrict__ Wh,     // W_hh (raw or packed)
                     const float* __restrict__ b_h,    // [H]
                     const float* __restrict__ h_in,   // [B,H]
                     float* __restrict__ h_out,        // [B,H]
                     int t)
{
    __shared__ float lds[32 * LDS_PITCH];

    const int tid  = threadIdx.x;
    const int wave = tid >> 5;
    const int lane = tid & 31;
    const int lm   = lane & 15;     // M (A/C) or N (B) index within a 16x16 tile
    const int lh   = lane >> 4;     // K-pair select: 0 -> {0,1}, 1 -> {2,3}

    const int m0 = blockIdx.y * 32;                       // batch-row base
    const int n  = blockIdx.x * 64 + wave * 16 + lm;      // this lane's output col

    v8f acc0 = {};   // rows m0 +  0..15
    v8f acc1 = {};   // rows m0 + 16..31

    // cooperative A-tile load: 128 threads x 16 floats = 32x64 elements
    const int arow = tid >> 2;         // 0..31
    const int acol = (tid & 3) * 16;   // 64B contiguous per thread

    const int a_off0 = lm * LDS_PITCH;
    const int a_off1 = (lm + 16) * LDS_PITCH;

    // per-lane packed-B base offset: element (k, n) pair -> k*H + 4n + 2lh
    const size_t bpack = (size_t)4 * n + 2 * lh;

    // ---------------- phase 1: K over I, A = x[:, t, :], B = W_hx -------------
    for (int k = 0; k < I_; k += KCHUNK) {
        const float* src = x + ((size_t)(m0 + arow) * T_ + t) * I_ + k + acol;
        float4 a0 = ((const float4*)src)[0];
        float4 a1 = ((const float4*)src)[1];
        float4 a2 = ((const float4*)src)[2];
        float4 a3 = ((const float4*)src)[3];
        float* dst = &lds[arow * LDS_PITCH + acol];
        ((float4*)dst)[0] = a0; ((float4*)dst)[1] = a1;
        ((float4*)dst)[2] = a2; ((float4*)dst)[3] = a3;
        __syncthreads();
        #pragma unroll
        for (int kk = 0; kk < KCHUNK; kk += 4) {
            const int ko = kk + 2 * lh;
            v2f af0 = *(const v2f*)&lds[a_off0 + ko];
            v2f af1 = *(const v2f*)&lds[a_off1 + ko];
            v2f b;
            if (PACKED) {
                b = *(const v2f*)(Wx + (size_t)(k + kk) * H_ + bpack);
            } else {
                const float* wp = Wx + (size_t)(k + ko) * H_ + n;
                b.x = wp[0]; b.y = wp[H_];
            }
            acc0 = __builtin_amdgcn_wmma_f32_16x16x4_f32(false, af0, false, b,
                                                         (short)0, acc0, false, false);
            acc1 = __builtin_amdgcn_wmma_f32_16x16x4_f32(false, af1, false, b,
                                                         (short)0, acc1, false, false);
        }
        __syncthreads();
    }

    // ---------------- phase 2: K over H, A = h_in, B = W_hh -------------------
    for (int k = 0; k < H_; k += KCHUNK) {
        const float* src = h_in + (size_t)(m0 + arow) * H_ + k + acol;
        float4 a0 = ((const float4*)src)[0];
        float4 a1 = ((const float4*)src)[1];
        float4 a2 = ((const float4*)src)[2];
        float4 a3 = ((const float4*)src)[3];
        float* dst = &lds[arow * LDS_PITCH + acol];
        ((float4*)dst)[0] = a0; ((float4*)dst)[1] = a1;
        ((float4*)dst)[2] = a2; ((float4*)dst)[3] = a3;
        __syncthreads();
        #pragma unroll
        for (int kk = 0; kk < KCHUNK; kk += 4) {
            const int ko = kk + 2 * lh;
            v2f af0 = *(const v2f*)&lds[a_off0 + ko];
            v2f af1 = *(const v2f*)&lds[a_off1 + ko];
            v2f b;
            if (PACKED) {
                b = *(const v2f*)(Wh + (size_t)(k + kk) * H_ + bpack);
            } else {
                const float* wp = Wh + (size_t)(k + ko) * H_ + n;
                b.x = wp[0]; b.y = wp[H_];
            }
            acc0 = __builtin_amdgcn_wmma_f32_16x16x4_f32(false, af0, false, b,
                                                         (short)0, acc0, false, false);
            acc1 = __builtin_amdgcn_wmma_f32_16x16x4_f32(false, af1, false, b,
                                                         (short)0, acc1, false, false);
        }
        __syncthreads();
    }

    // ---------------- epilogue: bias + tanh, scatter per C/D layout -----------
    const float bh = b_h[n];
    #pragma unroll
    for (int v = 0; v < 8; ++v) {
        const int row = m0 + v + 8 * lh;   // VGPR v: lanes 0-15 -> M=v, 16-31 -> M=v+8
        h_out[(size_t)row * H_ + n]        = tanhf(acc0[v] + bh);
        h_out[(size_t)(row + 16) * H_ + n] = tanhf(acc1[v] + bh);
    }
}

// ---------------------------------------------------------------------------
// Head: softmax(h_T @ W_ph + b_o). One block per batch element.
// ---------------------------------------------------------------------------
__global__ __launch_bounds__(256)
void rnn_head_kernel(const float* __restrict__ h,     // [B,H]
                     const float* __restrict__ W_ph,  // [H,O]
                     const float* __restrict__ b_o,   // [O]
                     float* __restrict__ out)         // [B,O]
{
    __shared__ float red[O_][256];
    const int b   = blockIdx.x;
    const int tid = threadIdx.x;

    float part[O_];
    #pragma unroll
    for (int o = 0; o < O_; ++o) part[o] = 0.f;

    for (int k = tid; k < H_; k += 256) {
        const float hv = h[(size_t)b * H_ + k];
        const float* w = W_ph + (size_t)k * O_;
        #pragma unroll
        for (int o = 0; o < O_; ++o) part[o] += hv * w[o];
    }
    #pragma unroll
    for (int o = 0; o < O_; ++o) red[o][tid] = part[o];
    __syncthreads();

    for (int s = 128; s > 0; s >>= 1) {
        if (tid < s) {
            #pragma unroll
            for (int o = 0; o < O_; ++o) red[o][tid] += red[o][tid + s];
        }
        __syncthreads();
    }

    if (tid == 0) {
        float logit[O_], mx = -1e30f;
        #pragma unroll
        for (int o = 0; o < O_; ++o) { logit[o] = red[o][0] + b_o[o]; mx = fmaxf(mx, logit[o]); }
        float sum = 0.f;
        #pragma unroll
        for (int o = 0; o < O_; ++o) { logit[o] = expf(logit[o] - mx); sum += logit[o]; }
        const float inv = 1.f / sum;
        #pragma unroll
        for (int o = 0; o < O_; ++o) out[(size_t)b * O_ + o] = logit[o] * inv;
    }
}

__global__ void zero_kernel(float* __restrict__ p, int n) {
    const int i = blockIdx.x * blockDim.x + threadIdx.x;
    if (i < n) p[i] = 0.f;
}

// ---------------------------------------------------------------------------
extern "C" void kernel_launch(void* const* d_in, const int* in_sizes, int n_in,
                              void* d_out, int out_size, void* d_ws, size_t ws_size,
                              hipStream_t stream) {
    const float* x    = (const float*)d_in[0];
    const float* W_hx = (const float*)d_in[1];
    const float* W_hh = (const float*)d_in[2];
    const float* b_h  = (const float*)d_in[3];
    const float* W_ph = (const float*)d_in[4];
    const float* b_o  = (const float*)d_in[5];
    float* out = (float*)d_out;

    const size_t BH = (size_t)B_ * H_;
    float* h0 = (float*)d_ws;                  // ping
    float* h1 = h0 + BH;                       // pong
    float* Wt_hx = h1 + BH;                    // packed weights (optional)
    float* Wt_hh = Wt_hx + (size_t)I_ * H_;

    const size_t need = (2 * BH + (size_t)I_ * H_ + (size_t)H_ * H_) * sizeof(float);
    const bool packed = (ws_size >= need);

    zero_kernel<<<(B_ * H_ + 255) / 256, 256, 0, stream>>>(h0, B_ * H_);

    if (packed) {
        const int gx = (I_ / 4) * H_;
        const int gh = (H_ / 4) * H_;
        repack_kernel<<<(gx + 255) / 256, 256, 0, stream>>>(W_hx, Wt_hx, I_, H_);
        repack_kernel<<<(gh + 255) / 256, 256, 0, stream>>>(W_hh, Wt_hh, H_, H_);
    }

    dim3 grid(H_ / 64, B_ / 32);               // 32 x 4 = 128 blocks, 512 waves
    for (int t = 0; t < T_; ++t) {
        const float* hin  = (t & 1) ? h1 : h0;
        float*       hout = (t & 1) ? h0 : h1;
        if (packed) {
            rnn_step_kernel<true><<<grid, 128, 0, stream>>>(
                x, Wt_hx, Wt_hh, b_h, hin, hout, t);
        } else {
            rnn_step_kernel<false><<<grid, 128, 0, stream>>>(
                x, W_hx, W_hh, b_h, hin, hout, t);
        }
    }
    // t = 511 (odd) wrote h0 -> final hidden state lives in h0
    rnn_head_kernel<<<B_, 256, 0, stream>>>(h0, W_ph, b_o, out);
}